// ProtoLayer_24790551232582
// MI455X (gfx1250) — compile-verified
//
#include <hip/hip_runtime.h>

typedef float v2f __attribute__((ext_vector_type(2)));
typedef float v4f __attribute__((ext_vector_type(4)));
typedef float v8f __attribute__((ext_vector_type(8)));

#define N_ROWS 65536
#define P_ROWS 512
#define DIM    256

#define MT  64     // M tile (feature rows) per block
#define PT  64     // P tile (prototype cols) per block iteration
#define KT  32     // K chunk staged in LDS (smaller -> double buffer fits)
#define LDT 36     // padded tile row stride (bank-conflict-free b64 frag loads)
#define LDS_STG 68 // padded stage row stride

// ---------------------------------------------------------------------------
// CDNA5 async global->LDS copy (GLOBAL_LOAD_ASYNC_TO_LDS_B128, ASYNCcnt).
// vdst = 32-bit LDS byte offset (low 32 bits of generic shared address),
// vaddr = 64-bit global address (VGPR pair).
// ---------------------------------------------------------------------------
__device__ __forceinline__ void async_copy_b128(const float* g, const float* l) {
  unsigned lds_off = (unsigned)(size_t)l;
  asm volatile("global_load_async_to_lds_b128 %0, %1, off"
               :
               : "v"(lds_off), "v"(g)
               : "memory");
}

// ---------------------------------------------------------------------------
// Kernel 1: squared row norms of F (-> ws[0..N)) and P (-> ws[N..N+P))
// ---------------------------------------------------------------------------
__global__ __launch_bounds__(256) void proto_norms_kernel(
    const float* __restrict__ F, const float* __restrict__ P,
    float* __restrict__ ws) {
  int t = blockIdx.x * blockDim.x + threadIdx.x;
  const float* src;
  float* dst;
  int row;
  if (t < N_ROWS) {
    src = F; row = t; dst = ws;
  } else if (t < N_ROWS + P_ROWS) {
    src = P; row = t - N_ROWS; dst = ws + N_ROWS;
  } else {
    return;
  }
  const v4f* p = (const v4f*)(src + (size_t)row * DIM);
  float s = 0.0f;
#pragma unroll 8
  for (int i = 0; i < DIM / 4; ++i) {
    v4f v = p[i];
    s += v.x * v.x + v.y * v.y + v.z * v.z + v.w * v.w;
  }
  dst[row] = s;
}

// ---------------------------------------------------------------------------
// Kernel 2: fp32 WMMA GEMM + distance epilogue, double-buffered async tiles.
//   out1[n*P + p] = fsq[n] + psq[p] - 2 * (F @ P^T)[n,p]
//   out2[p*N + n] = same value transposed, staged through LDS
// tile[buf]: rows 0..63 = F chunk, rows 64..127 = prototypes chunk
// ---------------------------------------------------------------------------
__global__ __launch_bounds__(256) void proto_dist_kernel(
    const float* __restrict__ F, const float* __restrict__ P,
    const float* __restrict__ fsq, const float* __restrict__ psq,
    float* __restrict__ out1, float* __restrict__ out2) {
  __shared__ float tile[2][128][LDT];
  __shared__ float stage[PT][LDS_STG];
  __shared__ float sFsq[MT];
  __shared__ float sPsq[PT];

  const int t    = threadIdx.x;
  const int lane = t & 31;
  const int wave = t >> 5;     // 0..7
  const int wcol = wave & 3;   // which 16-wide P column group
  const int wrow = wave >> 2;  // which 32-row M group
  const int g    = lane >> 4;  // K half-group (ISA layout: lane holds K=2g,2g+1)
  const int ml   = lane & 15;
  const int m0   = blockIdx.x * MT;

  if (t < MT) sFsq[t] = fsq[m0 + t];

  // async-copy indexing: 64 rows x 2 b128 slots per row quarter
  const int lrow = t >> 2;          // 0..63
  const int lcol = (t & 3) * 4;     // float col 0,4,8,12 (+16 for 2nd slot)

  for (int p0 = 0; p0 < P_ROWS; p0 += PT) {
    v8f acc0 = {};
    v8f acc1 = {};
    if (t < PT) sPsq[t] = psq[p0 + t];

    // issue chunk 0 into buffer 0
    {
      const float* gF = &F[(size_t)(m0 + lrow) * DIM + lcol];
      const float* gP = &P[(size_t)(p0 + lrow) * DIM + lcol];
      async_copy_b128(gF,      &tile[0][lrow][lcol]);
      async_copy_b128(gF + 16, &tile[0][lrow][lcol + 16]);
      async_copy_b128(gP,      &tile[0][64 + lrow][lcol]);
      async_copy_b128(gP + 16, &tile[0][64 + lrow][lcol + 16]);
    }

    for (int c = 0; c < DIM / KT; ++c) {
      const int buf = c & 1;
      if (c < DIM / KT - 1) {
        // prefetch chunk c+1 into the other buffer (safe: all reads of it
        // finished before the trailing barrier of iteration c-1)
        const int k0 = (c + 1) * KT;
        const float* gF = &F[(size_t)(m0 + lrow) * DIM + k0 + lcol];
        const float* gP = &P[(size_t)(p0 + lrow) * DIM + k0 + lcol];
        async_copy_b128(gF,      &tile[buf ^ 1][lrow][lcol]);
        async_copy_b128(gF + 16, &tile[buf ^ 1][lrow][lcol + 16]);
        async_copy_b128(gP,      &tile[buf ^ 1][64 + lrow][lcol]);
        async_copy_b128(gP + 16, &tile[buf ^ 1][64 + lrow][lcol + 16]);
        asm volatile("s_wait_asynccnt 0x4" ::: "memory");  // chunk c landed
      } else {
        asm volatile("s_wait_asynccnt 0x0" ::: "memory");
      }
      __syncthreads();  // all waves' chunk-c async loads visible in LDS

      const float(*fT)[LDT] = tile[buf];
      const float(*pT)[LDT] = tile[buf] + 64;
#pragma unroll
      for (int k = 0; k < KT; k += 4) {
        v2f a0 = *(const v2f*)&fT[wrow * 32 + ml][k + 2 * g];
        v2f a1 = *(const v2f*)&fT[wrow * 32 + 16 + ml][k + 2 * g];
        v2f b  = *(const v2f*)&pT[wcol * 16 + ml][k + 2 * g];
        acc0 = __builtin_amdgcn_wmma_f32_16x16x4_f32(
            false, a0, false, b, (short)0, acc0, false, false);
        acc1 = __builtin_amdgcn_wmma_f32_16x16x4_f32(
            false, a1, false, b, (short)0, acc1, false, false);
      }
      __syncthreads();  // reads of tile[buf] done before it is refilled
    }

    // Epilogue: C layout => VGPR j holds (M=j, lanes 0-15), (M=j+8, lanes
    // 16-31); N = lane%16. out1 written directly (coalesced 64B runs per
    // half-wave); transpose staged in LDS for out2.
    const int   colL = wcol * 16 + ml;
    const int   colG = p0 + colL;
    const float pv   = sPsq[colL];
#pragma unroll
    for (int j = 0; j < 8; ++j) {
      int r0 = wrow * 32 + j + 8 * g;
      float v0 = sFsq[r0] + pv - 2.0f * acc0[j];
      __builtin_nontemporal_store(v0, &out1[(size_t)(m0 + r0) * P_ROWS + colG]);
      stage[colL][r0] = v0;

      int r1 = r0 + 16;
      float v1 = sFsq[r1] + pv - 2.0f * acc1[j];
      __builtin_nontemporal_store(v1, &out1[(size_t)(m0 + r1) * P_ROWS + colG]);
      stage[colL][r1] = v1;
    }
    __syncthreads();

    // out2: each thread streams 64B contiguous; 4 threads cover 256B per row
    {
      int pr = t >> 2;
      int q  = (t & 3) * 16;
      const float* srcRow = &stage[pr][q];
      float* dst = &out2[(size_t)(p0 + pr) * N_ROWS + m0 + q];
#pragma unroll
      for (int i = 0; i < 4; ++i) {
        v4f v = *(const v4f*)&srcRow[i * 4];
        __builtin_nontemporal_store(v, (v4f*)&dst[i * 4]);
      }
    }
  }
}

// ---------------------------------------------------------------------------
extern "C" void kernel_launch(void* const* d_in, const int* in_sizes, int n_in,
                              void* d_out, int out_size, void* d_ws, size_t ws_size,
                              hipStream_t stream) {
  const float* F = (const float*)d_in[0];   // [N, D]
  const float* P = (const float*)d_in[1];   // [P, D]
  float* out  = (float*)d_out;
  float* out1 = out;                              // [N, P]
  float* out2 = out + (size_t)N_ROWS * P_ROWS;    // [P, N]
  float* fsq  = (float*)d_ws;                     // [N]
  float* psq  = fsq + N_ROWS;                     // [P]

  int normThreads = N_ROWS + P_ROWS;
  proto_norms_kernel<<<(normThreads + 255) / 256, 256, 0, stream>>>(F, P, fsq);

  proto_dist_kernel<<<N_ROWS / MT, 256, 0, stream>>>(F, P, fsq, psq, out1, out2);
}